// GlobalRecurrenceOptimized_13297218748581
// MI455X (gfx1250) — compile-verified
//
#include <hip/hip_runtime.h>
#include <math.h>

#define Dv 512
#define Tv 8192
#define Bv 4
#define M_TOTAL (Bv * Tv)          // 32768 rows
#define L_CHUNK 256
#define N_CHUNK (Tv / L_CHUNK)     // 32 chunks per sequence
#define LDS_STRIDE 516             // 512 + 4 pad: rows shift 4 banks -> conflict-free A-frag reads

typedef __attribute__((ext_vector_type(2))) float v2f;
typedef __attribute__((ext_vector_type(8))) float v8f;

__device__ __forceinline__ float sigmoid_f(float x) { return 1.0f / (1.0f + __expf(-x)); }

// Async memory -> LDS tile stage (gfx1250 GLOBAL_LOAD_ASYNC_TO_LDS_B128, ASYNCcnt).
// Each lane supplies its own LDS byte offset, so the padded stride is preserved.
__device__ __forceinline__ void async_stage_tile(const float* __restrict__ gbase,
                                                 float* __restrict__ lds, int m0, int tid)
{
    for (int j = 0; j < 16; ++j) {
        int idx = j * 1024 + tid * 4;
        int r = idx >> 9, c = idx & 511;
        const float* gsrc = gbase + (long)(m0 + r) * Dv + c;
        unsigned lds_off = (unsigned)(size_t)(lds + r * LDS_STRIDE + c);
        asm volatile("global_load_async_to_lds_b128 %0, %1, off"
                     :: "v"(lds_off), "v"(gsrc) : "memory");
    }
    asm volatile("s_wait_asynccnt 0x0" ::: "memory");
}

// ---------------------------------------------------------------------------
// Kernel 1: fused dual GEMM.  u = silu(x @ w_in[:, :D]),  g = sigmoid(x @ w_gate + b)
// Each wave: one shared A fragment feeds 4 WMMA accumulators (2 cols x {u,gate}).
// Block tile 32x128, waves in 2x4 grid.
// ---------------------------------------------------------------------------
__global__ __launch_bounds__(256) void k1_gemm_gate_u(
    const float* __restrict__ x, const float* __restrict__ w_in,
    const float* __restrict__ w_gate, const float* __restrict__ b_gate,
    float* __restrict__ u_ws, float* __restrict__ g_ws)
{
    __shared__ float xs[32 * LDS_STRIDE];
    const int tid  = threadIdx.x;
    const int m0   = blockIdx.y * 32;
    const int cblk = blockIdx.x * 128;

    async_stage_tile(x, xs, m0, tid);
    __syncthreads();

    const int wave = tid >> 5, lane = tid & 31;
    const int laneLo = lane & 15, laneHi = lane >> 4;
    const int wr = wave >> 2, wc = wave & 3;       // 2x4 wave grid -> 32x128 block tile
    const int rowA = wr * 16 + laneLo;
    const int c0   = cblk + wc * 32;               // this wave: cols [c0, c0+32)
    const float bg0 = b_gate[c0 + laneLo];
    const float bg1 = b_gate[c0 + 16 + laneLo];

    v8f accU0 = {}, accU1 = {}, accG0 = {}, accG1 = {};
    for (int k = 0; k < Dv; k += 4) {
        // A frag: ISA 16x4 layout (lanes0-15: K=k,k+1 ; lanes16-31: K=k+2,k+3)
        v2f a = *(const v2f*)(xs + rowA * LDS_STRIDE + k + 2 * laneHi);
        const int kb = k + 2 * laneHi;
        v2f bu0, bu1, bg0v, bg1v;
        bu0.x  = w_in  [(long)(kb + 0) * (2 * Dv) + c0 + laneLo];
        bu0.y  = w_in  [(long)(kb + 1) * (2 * Dv) + c0 + laneLo];
        bu1.x  = w_in  [(long)(kb + 0) * (2 * Dv) + c0 + 16 + laneLo];
        bu1.y  = w_in  [(long)(kb + 1) * (2 * Dv) + c0 + 16 + laneLo];
        bg0v.x = w_gate[(long)(kb + 0) * Dv       + c0 + laneLo];
        bg0v.y = w_gate[(long)(kb + 1) * Dv       + c0 + laneLo];
        bg1v.x = w_gate[(long)(kb + 0) * Dv       + c0 + 16 + laneLo];
        bg1v.y = w_gate[(long)(kb + 1) * Dv       + c0 + 16 + laneLo];
        if (k + 16 < Dv) {
            __builtin_prefetch(&w_in  [(long)(kb + 16) * (2 * Dv) + c0 + laneLo], 0, 0);
            __builtin_prefetch(&w_gate[(long)(kb + 16) * Dv       + c0 + laneLo], 0, 0);
        }
        accU0 = __builtin_amdgcn_wmma_f32_16x16x4_f32(false, a, false, bu0,  (short)0, accU0, false, false);
        accU1 = __builtin_amdgcn_wmma_f32_16x16x4_f32(false, a, false, bu1,  (short)0, accU1, false, false);
        accG0 = __builtin_amdgcn_wmma_f32_16x16x4_f32(false, a, false, bg0v, (short)0, accG0, false, false);
        accG1 = __builtin_amdgcn_wmma_f32_16x16x4_f32(false, a, false, bg1v, (short)0, accG1, false, false);
    }

    // epilogue: activations fused, C/D layout VGPR v -> M = v + 8*laneHi, N = laneLo
    for (int v = 0; v < 8; ++v) {
        int row = m0 + wr * 16 + v + 8 * laneHi;
        float u0 = accU0[v], u1 = accU1[v];
        float g0 = accG0[v] + bg0, g1 = accG1[v] + bg1;
        u_ws[(long)row * Dv + c0 + laneLo]      = u0 * sigmoid_f(u0);
        u_ws[(long)row * Dv + c0 + 16 + laneLo] = u1 * sigmoid_f(u1);
        g_ws[(long)row * Dv + c0 + laneLo]      = sigmoid_f(g0);
        g_ws[(long)row * Dv + c0 + 16 + laneLo] = sigmoid_f(g1);
    }
}

// ---------------------------------------------------------------------------
// Kernel 2: per-(b,chunk,d) local scan aggregates  (A = prod g,  S = local scan end)
// ---------------------------------------------------------------------------
__global__ __launch_bounds__(256) void k2_chunk_local(
    const float* __restrict__ g_ws, const float* __restrict__ u_ws,
    float* __restrict__ cA, float* __restrict__ cS)
{
    int g = blockIdx.x * blockDim.x + threadIdx.x;       // B*NC*D threads
    int d = g & (Dv - 1);
    int c = (g >> 9) & (N_CHUNK - 1);
    int b = g >> 14;
    long base = ((long)b * Tv + (long)c * L_CHUNK) * Dv + d;
    float A = 1.0f, S = 0.0f;
    for (int t = 0; t < L_CHUNK; ++t) {
        float gv = g_ws[base + (long)t * Dv];
        float uv = u_ws[base + (long)t * Dv];
        A *= gv;
        S = gv * S + uv;
    }
    cA[g] = A;
    cS[g] = S;
}

// ---------------------------------------------------------------------------
// Kernel 3: sequential exclusive prefix across the 32 chunks per (b,d) channel
// ---------------------------------------------------------------------------
__global__ __launch_bounds__(256) void k3_prefix(
    const float* __restrict__ cA, const float* __restrict__ cS,
    float* __restrict__ carry)
{
    int g = blockIdx.x * blockDim.x + threadIdx.x;       // B*D threads
    int d = g & (Dv - 1);
    int b = g >> 9;
    float h = 0.0f;
    for (int c = 0; c < N_CHUNK; ++c) {
        int idx = (b * N_CHUNK + c) * Dv + d;
        carry[idx] = h;
        h = cA[idx] * h + cS[idx];
    }
}

// ---------------------------------------------------------------------------
// Kernel 4: replay local scan with carry-in; h overwrites u buffer in place
// ---------------------------------------------------------------------------
__global__ __launch_bounds__(256) void k4_apply(
    const float* __restrict__ g_ws, const float* __restrict__ carry,
    float* __restrict__ hu_ws)
{
    int g = blockIdx.x * blockDim.x + threadIdx.x;
    int d = g & (Dv - 1);
    int c = (g >> 9) & (N_CHUNK - 1);
    int b = g >> 14;
    long base = ((long)b * Tv + (long)c * L_CHUNK) * Dv + d;
    float h = carry[g];
    for (int t = 0; t < L_CHUNK; ++t) {
        long idx = base + (long)t * Dv;
        float gv = g_ws[idx];
        float uv = hu_ws[idx];
        h = gv * h + uv;
        hu_ws[idx] = h;
    }
}

// ---------------------------------------------------------------------------
// Kernel 5: RMS-norm fused into A-tile staging, then WMMA GEMM with w_out.
// Each wave: one shared A fragment, two column-tile accumulators.
// ---------------------------------------------------------------------------
__global__ __launch_bounds__(256) void k5_gemm_out(
    const float* __restrict__ h_ws, const float* __restrict__ norm_w,
    const float* __restrict__ w_out, float* __restrict__ out)
{
    __shared__ float xs[32 * LDS_STRIDE];
    __shared__ float psum[32 * 8];
    __shared__ float rscale[32];
    const int tid  = threadIdx.x;
    const int m0   = blockIdx.y * 32;
    const int cblk = blockIdx.x * 128;

    async_stage_tile(h_ws, xs, m0, tid);
    __syncthreads();

    // row-wise sum of squares: 8 segment-partials per row
    {
        int r = tid >> 3, seg = tid & 7;
        const float* p = xs + r * LDS_STRIDE + seg * 64;
        float s = 0.0f;
        for (int i = 0; i < 64; ++i) { float v = p[i]; s += v * v; }
        psum[r * 8 + seg] = s;
    }
    __syncthreads();
    if (tid < 32) {
        float s = 0.0f;
        for (int i = 0; i < 8; ++i) s += psum[tid * 8 + i];
        rscale[tid] = rsqrtf(s * (1.0f / Dv) + 1e-6f);
    }
    __syncthreads();
    // normalize in LDS: h * rsqrt(var) * norm_w  (norm_w folded per-column)
    for (int j = 0; j < 16; ++j) {
        int idx = j * 1024 + tid * 4;
        int r = idx >> 9, c = idx & 511;
        float sc = rscale[r];
        float* p = xs + r * LDS_STRIDE + c;
        p[0] *= sc * norm_w[c + 0];
        p[1] *= sc * norm_w[c + 1];
        p[2] *= sc * norm_w[c + 2];
        p[3] *= sc * norm_w[c + 3];
    }
    __syncthreads();

    const int wave = tid >> 5, lane = tid & 31;
    const int laneLo = lane & 15, laneHi = lane >> 4;
    const int wr = wave >> 2, wc = wave & 3;
    const int rowA = wr * 16 + laneLo;
    const int c0   = cblk + wc * 32;

    v8f acc0 = {}, acc1 = {};
    for (int k = 0; k < Dv; k += 4) {
        v2f a = *(const v2f*)(xs + rowA * LDS_STRIDE + k + 2 * laneHi);
        const int kb = k + 2 * laneHi;
        v2f b0, b1;
        b0.x = w_out[(long)(kb + 0) * Dv + c0 + laneLo];
        b0.y = w_out[(long)(kb + 1) * Dv + c0 + laneLo];
        b1.x = w_out[(long)(kb + 0) * Dv + c0 + 16 + laneLo];
        b1.y = w_out[(long)(kb + 1) * Dv + c0 + 16 + laneLo];
        if (k + 16 < Dv)
            __builtin_prefetch(&w_out[(long)(kb + 16) * Dv + c0 + laneLo], 0, 0);
        acc0 = __builtin_amdgcn_wmma_f32_16x16x4_f32(false, a, false, b0, (short)0, acc0, false, false);
        acc1 = __builtin_amdgcn_wmma_f32_16x16x4_f32(false, a, false, b1, (short)0, acc1, false, false);
    }
    for (int v = 0; v < 8; ++v) {
        int row = m0 + wr * 16 + v + 8 * laneHi;
        out[(long)row * Dv + c0 + laneLo]      = acc0[v];
        out[(long)row * Dv + c0 + 16 + laneLo] = acc1[v];
    }
}

// ---------------------------------------------------------------------------
extern "C" void kernel_launch(void* const* d_in, const int* in_sizes, int n_in,
                              void* d_out, int out_size, void* d_ws, size_t ws_size,
                              hipStream_t stream)
{
    (void)in_sizes; (void)n_in; (void)out_size; (void)ws_size;
    const float* x      = (const float*)d_in[0];
    const float* w_in   = (const float*)d_in[1];
    const float* w_gate = (const float*)d_in[2];
    const float* b_gate = (const float*)d_in[3];
    const float* norm_w = (const float*)d_in[4];
    const float* w_out  = (const float*)d_in[5];
    float* out = (float*)d_out;

    const long BTD = (long)Bv * Tv * Dv;                 // 16,777,216 floats
    float* u_ws  = (float*)d_ws;                         // 64 MB (becomes h in place)
    float* g_ws  = u_ws + BTD;                           // 64 MB
    float* cA    = g_ws + BTD;                           // 256 KB
    float* cS    = cA + (long)Bv * N_CHUNK * Dv;         // 256 KB
    float* carry = cS + (long)Bv * N_CHUNK * Dv;         // 256 KB

    dim3 blk(256);
    dim3 gridGemm(Dv / 128, M_TOTAL / 32);               // 4 x 1024 blocks

    k1_gemm_gate_u<<<gridGemm, blk, 0, stream>>>(x, w_in, w_gate, b_gate, u_ws, g_ws);
    k2_chunk_local<<<(Bv * N_CHUNK * Dv) / 256, blk, 0, stream>>>(g_ws, u_ws, cA, cS);
    k3_prefix     <<<(Bv * Dv) / 256,            blk, 0, stream>>>(cA, cS, carry);
    k4_apply      <<<(Bv * N_CHUNK * Dv) / 256, blk, 0, stream>>>(g_ws, carry, u_ws);
    k5_gemm_out   <<<gridGemm, blk, 0, stream>>>(u_ws, norm_w, w_out, out);
}